// GaussianImage_Cholesky_89670327206144
// MI455X (gfx1250) — compile-verified
//
#include <hip/hip_runtime.h>
#include <math.h>

// ---------------------------------------------------------------------------
// GaussianImage (Cholesky) renderer for gfx1250.
//   H = W = 256, 256 curves x 64 gaussians = 16384 gaussians, all composited
//   per pixel in depth order.  Hot loop = 1.07e9 (pixel,gaussian) pairs.
//   Power term computed as a 6-feature bilinear form via V_WMMA_F32_16X16X4_F32.
// ---------------------------------------------------------------------------

typedef __attribute__((ext_vector_type(2))) float v2f;
typedef __attribute__((ext_vector_type(8))) float v8f;

#define NG        16384      // total gaussians
#define GT        256        // gaussians per LDS super-tile
#define NSUPER    (NG / GT)  // 64
#define L2E       1.4426950408889634f

#if __has_builtin(__builtin_amdgcn_exp2f)
#define EXP2F(x) __builtin_amdgcn_exp2f(x)
#else
#define EXP2F(x) exp2f(x)
#endif

__device__ __forceinline__ float sigm(float x) { return 1.0f / (1.0f + expf(-x)); }

// ---------------------------------------------------------------------------
// Preprocess: one thread per gaussian (64 blocks x 256).
//  - stable depth rank of its curve (LDS, 256 compares)
//  - conic -> pre-scaled quadratic coefficients q0,q1,q2 (include -0.5 and log2e)
//  - sigmoid(color), sigmoid(opacity)
//  - Bezier/Bernstein evaluation -> tanh -> pixel-space mean
//  Writes SoA arrays in *sorted* order:
//    ga[g] = (q0, q1, q2, mean_x), gb[g] = (mean_y, R, G, B), gal[g] = alpha
// ---------------------------------------------------------------------------
__global__ __launch_bounds__(256) void gi_prep(
    const float* __restrict__ ctrl,   // (256,10,2)
    const float* __restrict__ fdc,    // (256,3)
    const float* __restrict__ chol,   // (256,3)
    const float* __restrict__ opac,   // (256,1)
    const float* __restrict__ depth,  // (256,1)
    float4* __restrict__ ga, float4* __restrict__ gb, float* __restrict__ gal)
{
  __shared__ float sd[256];
  const int t = threadIdx.x;
  sd[t] = depth[t];
  __syncthreads();

  const int tid = blockIdx.x * 256 + t;   // gaussian id in original order
  const int c = tid >> 6;                 // curve
  const int j = tid & 63;                 // gaussian within curve (k*32+s)

  // stable rank of this curve's depth
  const float dc = sd[c];
  int rank = 0;
  for (int i = 0; i < 256; ++i) {
    const float di = sd[i];
    rank += (di < dc || (di == dc && i < c)) ? 1 : 0;
  }
  const int slot = rank * 64 + j;

  // conic from cholesky
  const float c0 = chol[c * 3 + 0] + 0.5f;
  const float c1 = chol[c * 3 + 1];
  const float c2 = chol[c * 3 + 2] + 0.5f;
  const float s00 = c0 * c0, s01 = c0 * c1, s11 = c1 * c1 + c2 * c2;
  const float inv = 1.0f / (s00 * s11 - s01 * s01);
  const float A = s11 * inv;      // conic[0]  (dx^2)
  const float Bc = -s01 * inv;    // conic[1]  (dx*dy)
  const float C = s00 * inv;      // conic[2]  (dy^2)
  const float q0 = -0.5f * L2E * A;
  const float q1 = -0.5f * L2E * C;
  const float q2 = -L2E * Bc;

  const float Rcol = sigm(fdc[c * 3 + 0]);
  const float Gcol = sigm(fdc[c * 3 + 1]);
  const float Bcol = sigm(fdc[c * 3 + 2]);
  const float al   = sigm(opac[c]);

  // Bernstein weights at sample s (deg 5), segment k
  const int k = j >> 5, s = j & 31;
  const float tt = 0.007f + (float)s * (0.986f / 31.0f);
  const float u = 1.0f - tt;
  const float u2 = u * u, u3 = u2 * u, u4 = u2 * u2, u5 = u4 * u;
  const float t2 = tt * tt, t3 = t2 * tt, t4 = t2 * t2, t5 = t4 * tt;
  float bw[6];
  bw[0] = u5;             bw[1] = 5.0f * tt * u4;  bw[2] = 10.0f * t2 * u3;
  bw[3] = 10.0f * t3 * u2; bw[4] = 5.0f * t4 * u;  bw[5] = t5;

  float px = 0.0f, py = 0.0f;
#pragma unroll
  for (int jj = 0; jj < 6; ++jj) {
    int ii = k * 5 + jj;            // seg_idx = (k*5+jj) % 10
    if (ii >= 10) ii -= 10;
    const float cx = ctrl[(c * 10 + ii) * 2 + 0];
    const float cy = ctrl[(c * 10 + ii) * 2 + 1];
    px += bw[jj] * cx;
    py += bw[jj] * cy;
  }
  const float mx = (tanhf(px) * 0.5f + 0.5f) * 256.0f;
  const float my = (tanhf(py) * 0.5f + 0.5f) * 256.0f;

  ga[slot] = make_float4(q0, q1, q2, mx);
  gb[slot] = make_float4(my, Rcol, Gcol, Bcol);
  gal[slot] = al;
}

// ---------------------------------------------------------------------------
// Render: 512 blocks x 256 threads.  Block = 16(x) x 8(y) pixel tile,
// wave w owns row w (16 pixels, N-dim of WMMA).  M-dim = 16 gaussians.
// power*log2e = A(16g x 6feat) @ B(6feat x 16px) via two f32 16x16x4 WMMAs.
// D layout: lane = pixel (n = lane&15), VGPR r = gaussian r (+8 upper half).
// Each half composites its 8-gaussian segment, halves merged via shfl_xor(16)
// using the associativity of front-to-back compositing.
// ---------------------------------------------------------------------------
__global__ __launch_bounds__(256) void gi_render(
    const float4* __restrict__ ga, const float4* __restrict__ gb,
    const float* __restrict__ gal, const float* __restrict__ bg,
    float* __restrict__ out)
{
  __shared__ float2 s_h01[GT];   // (h0,h1) = (q0,q1)
  __shared__ float2 s_h23[GT];   // (h2,h3)
  __shared__ float2 s_h45[GT];   // (h4,h5)
  __shared__ float4 s_ca[GT];    // (R,G,B,alpha)

  const int t    = threadIdx.x;
  const int lane = t & 31;
  const int wv   = t >> 5;            // wave id = row in tile (0..7)
  const int n    = lane & 15;         // pixel column in tile
  const bool hi  = (lane & 16) != 0;  // upper half-wave

  const int b  = blockIdx.x;
  const int tx = (b & 15) << 4;       // tile origin x
  const int ty = (b >> 4) << 3;       // tile origin y
  const float ox = (float)tx + 8.0f;  // tile center (pixel coords are +0.5)
  const float oy = (float)ty + 4.0f;

  // tile-centered pixel coords (small -> f32-exact bilinear form)
  const float pxc = (float)n - 7.5f;
  const float pyc = (float)wv - 3.5f;

  // B matrix (K x N = features x pixels), per-lane constants:
  //  VGPR0: K0|K2, VGPR1: K1|K3 ; second k-step: K4|K6(=0), K5|K7(=0)
  v2f b_lo, b_hi;
  b_lo.x = hi ? pxc * pyc : pxc * pxc;   // f2 | f0
  b_lo.y = hi ? pxc       : pyc * pyc;   // f3 | f1
  b_hi.x = hi ? 0.0f      : pyc;         // 0  | f4
  b_hi.y = hi ? 0.0f      : 1.0f;        // 0  | f5

  float accR = 0.0f, accG = 0.0f, accB = 0.0f, T = 1.0f;

#pragma unroll 1
  for (int st = 0; st < NSUPER; ++st) {
    __syncthreads();
    // ---- stage 256 gaussians, recomputing tile-centered h3..h5 ----
    {
      const int g = st * GT + t;
      const float4 a0 = ga[g];
      const float4 a1 = gb[g];
      const float al  = gal[g];
      if (st + 1 < NSUPER) {
        __builtin_prefetch((const void*)(ga + g + GT), 0, 1);
        __builtin_prefetch((const void*)(gb + g + GT), 0, 1);
      }
      const float q0 = a0.x, q1 = a0.y, q2 = a0.z;
      const float mx = a0.w - ox;
      const float my = a1.x - oy;
      const float h3 = -(2.0f * q0 * mx + q2 * my);
      const float h4 = -(2.0f * q1 * my + q2 * mx);
      const float h5 = (q0 * mx) * mx + (q1 * my) * my + (q2 * mx) * my;
      s_h01[t] = make_float2(q0, q1);
      s_h23[t] = make_float2(q2, h3);
      s_h45[t] = make_float2(h4, h5);
      s_ca[t]  = make_float4(a1.y, a1.z, a1.w, al);
    }
    __syncthreads();

    // ---- 16 WMMA tiles of 16 gaussians each, in depth order ----
#pragma unroll 1
    for (int tt = 0; tt < 16; ++tt) {
      const int base = tt << 4;
      const int gl = base + n;
      // A matrix (M x K = gaussians x features):
      //  VGPR0: K0|K2, VGPR1: K1|K3 ; second k-step: K4|K6(=0), K5|K7(=0)
      const float2* hp = hi ? s_h23 : s_h01;
      const float2 alo2 = hp[gl];
      const float2 ahi2 = s_h45[gl];
      v2f a_lo, a_hi;
      a_lo.x = alo2.x;             a_lo.y = alo2.y;
      a_hi.x = hi ? 0.0f : ahi2.x; a_hi.y = hi ? 0.0f : ahi2.y;

      v8f acc = {0.0f, 0.0f, 0.0f, 0.0f, 0.0f, 0.0f, 0.0f, 0.0f};
      acc = __builtin_amdgcn_wmma_f32_16x16x4_f32(
          false, a_lo, false, b_lo, (short)0, acc, false, false);
      acc = __builtin_amdgcn_wmma_f32_16x16x4_f32(
          false, a_hi, false, b_hi, (short)0, acc, false, false);
      // acc[r] = power*log2e for gaussian base + r (+8 in upper half), pixel n

      // segment composite (8 gaussians of this half), local T starts at 1
      const int cbase = base + (hi ? 8 : 0);
      float sR = 0.0f, sG = 0.0f, sB = 0.0f, sT = 1.0f;
#pragma unroll
      for (int r = 0; r < 8; ++r) {
        const float4 ca = s_ca[cbase + r];
        const float e = EXP2F(acc[r]);
        const float a = fminf(ca.w * e, 0.999f);
        const float wgt = sT * a;
        sR += wgt * ca.x;
        sG += wgt * ca.y;
        sB += wgt * ca.z;
        sT *= (1.0f - a);
      }
      // merge half-waves (lower half = earlier gaussians)
      const float oR = __shfl_xor(sR, 16, 32);
      const float oG = __shfl_xor(sG, 16, 32);
      const float oB = __shfl_xor(sB, 16, 32);
      const float oT = __shfl_xor(sT, 16, 32);
      const float fR = hi ? oR : sR, fG = hi ? oG : sG;
      const float fB = hi ? oB : sB, fT = hi ? oT : sT;
      const float lR = hi ? sR : oR, lG = hi ? sG : oG;
      const float lB = hi ? sB : oB, lT = hi ? sT : oT;
      const float tR = fR + fT * lR;
      const float tG = fG + fT * lG;
      const float tB = fB + fT * lB;
      const float tT = fT * lT;
      // fold tile segment into running state (duplicated across halves)
      accR += T * tR;
      accG += T * tG;
      accB += T * tB;
      T *= tT;
    }
  }

  if (!hi) {
    const float b0 = bg[0], b1 = bg[1], b2 = bg[2];
    const int x = tx + n, y = ty + wv;
    const int o = (y * 256 + x) * 3;
    out[o + 0] = fminf(fmaxf(accR + T * b0, 0.0f), 1.0f);
    out[o + 1] = fminf(fmaxf(accG + T * b1, 0.0f), 1.0f);
    out[o + 2] = fminf(fmaxf(accB + T * b2, 0.0f), 1.0f);
  }
}

// ---------------------------------------------------------------------------
extern "C" void kernel_launch(void* const* d_in, const int* in_sizes, int n_in,
                              void* d_out, int out_size, void* d_ws, size_t ws_size,
                              hipStream_t stream) {
  (void)in_sizes; (void)n_in; (void)out_size; (void)ws_size;
  const float* ctrl  = (const float*)d_in[0];
  const float* fdc   = (const float*)d_in[1];
  const float* chol  = (const float*)d_in[2];
  const float* opac  = (const float*)d_in[3];
  const float* depth = (const float*)d_in[4];
  const float* bg    = (const float*)d_in[5];

  float4* ga  = (float4*)d_ws;         // 16384 * 16 B
  float4* gb  = ga + NG;               // 16384 * 16 B
  float*  gal = (float*)(gb + NG);     // 16384 * 4  B   (total ~576 KB)
  float*  out = (float*)d_out;

  gi_prep<<<NG / 256, 256, 0, stream>>>(ctrl, fdc, chol, opac, depth, ga, gb, gal);
  gi_render<<<512, 256, 0, stream>>>(ga, gb, gal, bg, out);
}